// TargetPositionPredictor_53051436040795
// MI455X (gfx1250) — compile-verified
//
#include <hip/hip_runtime.h>
#include <math.h>

typedef float v2f __attribute__((ext_vector_type(2)));
typedef float v8f __attribute__((ext_vector_type(8)));

#define LMAXV 4
#define MULV 128
#define NCH 4
#define NRAD 64
#define RBETA 90
#define RALPHA 180
#define NG 64
#define KDIM 25
#define NPTS (RBETA*RALPHA)        /* 16200 */
#define NT ((NPTS + 15)/16)        /* 1013 N-tiles */
#define NGR (NG*NRAD)              /* 4096 (g,r) rows */
#define MT (NGR/4)                 /* 1024 M-tiles (16 rows = 4 gr x 4 ch) */
#define PC_SIZE (NG*NCH*NRAD*KDIM) /* 409600 */
#define NT_CHUNKS 16
#define NT_CHUNK ((NT + NT_CHUNKS - 1)/NT_CHUNKS)  /* 64 */
#define NT_PAD (NT_CHUNKS*NT_CHUNK)                /* 1024 (zero padded) */

#define STAGE_NT 4                         /* N-tiles staged per LDS buffer fill */
#define TILE_F   (7*32*2)                  /* 448 floats per packed tile */
#define STAGE_F  (STAGE_NT*TILE_F)         /* 1792 floats / 7168 B per stage */
#define STAGE_B128 (STAGE_F/4)             /* 448 b128 transfers per stage */
#define NSTAGES (NT_CHUNK/STAGE_NT)        /* 16 */

/* workspace layout (float units) */
#define WS_YPACK   0
#define YPACK_F    (NT_PAD*TILE_F)          /* 458752 */
#define WS_APACK   (WS_YPACK + YPACK_F)
#define APACK_F    (MT*TILE_F)              /* 458752 */
#define WS_GMAX    (WS_APACK + APACK_F)     /* 64 uints */
#define WS_YNODES  (WS_GMAX + 64)
#define WS_ZERO_N  WS_YNODES                /* zero Ypack+Apack+gmax each call */

#define PI_D 3.14159265358979323846
#define PI_F 3.14159265358979323846f

__device__ __forceinline__ unsigned fenc(float f) {
    unsigned u = __float_as_uint(f);
    return (u & 0x80000000u) ? ~u : (u | 0x80000000u);
}
__device__ __forceinline__ float fdec(unsigned e) {
    return (e & 0x80000000u) ? __uint_as_float(e & 0x7FFFFFFFu)
                             : __uint_as_float(~e);
}

/* CDNA5 async global->LDS copy (ASYNCcnt-tracked).
   VDST VGPR = LDS byte offset (= low 32 bits of generic pointer to __shared__),
   VADDR = 64-bit global address, no SADDR. */
__device__ __forceinline__ void async_b128(float* lds_dst, const float* gsrc) {
    unsigned loff = (unsigned)(unsigned long long)lds_dst;
    unsigned long long ga = (unsigned long long)gsrc;
    asm volatile("global_load_async_to_lds_b128 %0, %1, off"
                 :: "v"(loff), "v"(ga) : "memory");
}
__device__ __forceinline__ void wait_async0() {
    asm volatile("s_wait_asynccnt 0x0" ::: "memory");
}

/* ---------- zero the pack buffers + per-graph max slots ---------- */
__global__ void k_init(float* ws) {
    int i = blockIdx.x * blockDim.x + threadIdx.x;
    if (i < WS_ZERO_N) ws[i] = 0.0f;
}

/* ---------- Gauss-Legendre nodes of degree 90 (Newton, fp64) ---------- */
__global__ void k_nodes(float* ws) {
    int b = threadIdx.x;
    if (b >= RBETA) return;
    const int n = RBETA;
    int k = (n - 1) - b;  /* descending guess -> ascending node order (numpy leggauss) */
    double x = cos(PI_D * (k + 0.75) / (n + 0.5));
    for (int it = 0; it < 50; ++it) {
        double p0 = 1.0, p1 = x;
        for (int j = 2; j <= n; ++j) {
            double p2 = ((2.0 * j - 1.0) * x * p1 - (j - 1.0) * p0) / j;
            p0 = p1; p1 = p2;
        }
        double dp = n * (x * p1 - p0) / (x * x - 1.0);
        x -= p1 / dp;
    }
    ws[WS_YNODES + b] = (float)x;
}

/* ---------- real spherical harmonics, packed for WMMA-B ----------
   slot k = 4*s + (half?2:0) + lo/hi matches V_WMMA_F32_16X16X4_F32 B layout */
__global__ void k_ybasis(float* ws) {
    int n = blockIdx.x * blockDim.x + threadIdx.x;
    if (n >= NPTS) return;  /* padded columns stay zero from k_init */
    int b = n / RALPHA, ia = n % RALPHA;
    float y = ws[WS_YNODES + b];
    float s = sqrtf(fmaxf(1.0f - y * y, 0.0f));

    float P[LMAXV + 1][LMAXV + 1];
    P[0][0] = sqrtf(1.0f / (4.0f * PI_F));
    for (int m = 1; m <= LMAXV; ++m)
        P[m][m] = -sqrtf((2.0f * m + 1.0f) / (2.0f * m)) * s * P[m - 1][m - 1];
    for (int m = 0; m < LMAXV; ++m)
        P[m + 1][m] = sqrtf(2.0f * m + 3.0f) * y * P[m][m];
    for (int m = 0; m <= LMAXV; ++m)
        for (int l = m + 2; l <= LMAXV; ++l) {
            float a = sqrtf((4.0f * l * l - 1.0f) / ((float)l * l - (float)m * m));
            float bb = sqrtf(((l - 1.0f) * (l - 1.0f) - (float)m * m) /
                             (4.0f * (l - 1.0f) * (l - 1.0f) - 1.0f));
            P[l][m] = a * (y * P[l - 1][m] - bb * P[l - 2][m]);
        }

    float alpha = 2.0f * PI_F * (float)ia / (float)RALPHA;
    int nt = n >> 4, j = n & 15;
    float* Yp = ws + WS_YPACK;
    const float SQ2 = 1.41421356237309515f;
    int idx = 0;
    for (int l = 0; l <= LMAXV; ++l)
        for (int m = -l; m <= l; ++m) {
            float v;
            if (m == 0)      v = P[l][0];
            else if (m > 0)  v = SQ2 * P[l][m] * cosf((float)m * alpha);
            else             v = SQ2 * P[l][-m] * sinf((float)(-m) * alpha);
            int kk = idx++;
            int slot = ((nt * 7 + (kk >> 2)) * 32 + ((kk & 2) ? 16 : 0) + j) * 2 + (kk & 1);
            Yp[slot] = v;
        }
}

/* ---------- embedding * equivariant linear -> position_coeffs + WMMA-A pack ---------- */
__global__ void __launch_bounds__(256)
k_coeffs(const float* __restrict__ x, const float* __restrict__ table,
         const int* __restrict__ species,
         const float* __restrict__ W0, const float* __restrict__ W1,
         const float* __restrict__ W2, const float* __restrict__ W3,
         const float* __restrict__ W4,
         float* __restrict__ pc_out, float* __restrict__ ws) {
    int g = blockIdx.x;
    int l = blockIdx.y;
    int d = 2 * l + 1;
    const float* Wl = (l == 0) ? W0 : (l == 1) ? W1 : (l == 2) ? W2 : (l == 3) ? W3 : W4;

    __shared__ float sh[MULV * 9];
    int sp = species[g];
    int tot = MULV * d;
    int xoff = g * (MULV * (LMAXV + 1) * (LMAXV + 1)) + MULV * l * l;
    for (int idx = threadIdx.x; idx < tot; idx += blockDim.x) {
        int i = idx / d;  /* multiplicity index */
        sh[idx] = x[xoff + idx] * table[sp * (MULV * (LMAXV + 1)) + l * MULV + i];
    }
    __syncthreads();

    int o = threadIdx.x;  /* 0..255 output mul */
    float acc[9];
    #pragma unroll
    for (int m = 0; m < 9; ++m) acc[m] = 0.0f;
    for (int i = 0; i < MULV; ++i) {
        float w = Wl[i * 256 + o];
        for (int m = 0; m < d; ++m) acc[m] += sh[i * d + m] * w;
    }

    const float inv = 0.08838834764831845f;  /* 1/sqrt(128) */
    int c = o >> 6, r = o & 63;
    int gr = g * NRAD + r;
    int t = gr >> 2;
    int mrow = (gr & 3) * 4 + c;  /* row within 16-row M-tile */
    float* Ap = ws + WS_APACK;
    for (int m = 0; m < d; ++m) {
        float v = acc[m] * inv;
        int kk = l * l + m;
        pc_out[((g * NCH + c) * NRAD + r) * KDIM + kk] = v;
        Ap[((t * 7 + (kk >> 2)) * 32 + ((kk & 2) ? 16 : 0) + mrow) * 2 + (kk & 1)] = v;
    }
}

/* ---------- fused s2grid GEMM (f32 WMMA) + channel logsumexp + per-graph max ----------
   8 waves/block share one N-chunk: Y tiles are staged once per block into LDS with
   CDNA5 async global->LDS copies (ASYNCcnt), double-buffered against the WMMA chain. */
__global__ void __launch_bounds__(256)
k_s2lse(float* __restrict__ ws, float* __restrict__ logits) {
    __shared__ float shY[2][STAGE_F];

    const float* Yp = ws + WS_YPACK;
    const float* Ap = ws + WS_APACK;
    unsigned* gmax = (unsigned*)(ws + WS_GMAX);

    int tid  = threadIdx.x;
    int lane = tid & 31;
    int wid  = tid >> 5;
    int t = blockIdx.y * 8 + wid;  /* M-tile 0..1023; all 4 gr rows share one graph */
    int g = t >> 4;
    int nt0 = blockIdx.x * NT_CHUNK;

    /* A fragments live in registers for the whole N sweep */
    const v2f* A2 = (const v2f*)Ap + t * 7 * 32 + lane;
    v2f a[7];
    #pragma unroll
    for (int s = 0; s < 7; ++s) a[s] = A2[s * 32];

    /* prefetch stage 0 */
    {
        const float* src = Yp + (size_t)nt0 * TILE_F;
        for (int i = tid; i < STAGE_B128; i += 256)
            async_b128(&shY[0][i * 4], src + i * 4);
    }
    wait_async0();
    __syncthreads();

    float wmax = -3.0e38f;
    for (int st = 0; st < NSTAGES; ++st) {
        int cur = st & 1;
        /* stream next stage into the other buffer while computing this one */
        if (st + 1 < NSTAGES) {
            const float* src = Yp + (size_t)(nt0 + (st + 1) * STAGE_NT) * TILE_F;
            for (int i = tid; i < STAGE_B128; i += 256)
                async_b128(&shY[cur ^ 1][i * 4], src + i * 4);
        }

        #pragma unroll
        for (int j = 0; j < STAGE_NT; ++j) {
            int nt = nt0 + st * STAGE_NT + j;
            const v2f* Bs = (const v2f*)(&shY[cur][0]) + j * (TILE_F / 2) + lane;
            v8f acc = {0.f, 0.f, 0.f, 0.f, 0.f, 0.f, 0.f, 0.f};
            #pragma unroll
            for (int s = 0; s < 7; ++s) {
                v2f bfrag = Bs[s * 32];   /* ds_load_b64, conflict-free */
                acc = __builtin_amdgcn_wmma_f32_16x16x4_f32(
                    false, a[s], false, bfrag, (short)0, acc, false, false);
            }
            /* lanes 0-15: acc[v]=row v (N=lane); lanes 16-31: row v+8 (N=lane-16).
               Rows group 4 channels -> in-lane logsumexp over acc[0..3], acc[4..7]. */
            float m0 = fmaxf(fmaxf(acc[0], acc[1]), fmaxf(acc[2], acc[3]));
            float l0 = m0 + __logf(__expf(acc[0] - m0) + __expf(acc[1] - m0) +
                                   __expf(acc[2] - m0) + __expf(acc[3] - m0));
            float m1 = fmaxf(fmaxf(acc[4], acc[5]), fmaxf(acc[6], acc[7]));
            float l1 = m1 + __logf(__expf(acc[4] - m1) + __expf(acc[5] - m1) +
                                   __expf(acc[6] - m1) + __expf(acc[7] - m1));
            int col = nt * 16 + (lane & 15);
            int grb = t * 4 + ((lane >> 4) << 1);
            if (col < NPTS) {
                logits[(size_t)grb * NPTS + col]       = l0;
                logits[(size_t)(grb + 1) * NPTS + col] = l1;
                wmax = fmaxf(wmax, fmaxf(l0, l1));
            }
        }

        wait_async0();       /* next stage landed in LDS */
        __syncthreads();     /* all waves done reading cur, data visible */
    }

    /* wave32 max reduction, one atomic per wave */
    #pragma unroll
    for (int o = 16; o > 0; o >>= 1)
        wmax = fmaxf(wmax, __shfl_xor(wmax, o, 32));
    if (lane == 0) atomicMax(&gmax[g], fenc(wmax));
}

/* ---------- subtract per-graph max (vectorized) ---------- */
__global__ void k_submax(float* __restrict__ logits, const float* __restrict__ ws) {
    const unsigned* gmax = (const unsigned*)(ws + WS_GMAX);
    const long long n4 = (long long)NG * NRAD * NPTS / 4;
    const int per_g4 = NRAD * NPTS / 4;
    for (long long i = blockIdx.x * (long long)blockDim.x + threadIdx.x; i < n4;
         i += (long long)gridDim.x * blockDim.x) {
        int g = (int)(i / per_g4);
        float mx = fdec(gmax[g]);
        float4 v = ((float4*)logits)[i];
        v.x -= mx; v.y -= mx; v.z -= mx; v.w -= mx;
        ((float4*)logits)[i] = v;
    }
}

extern "C" void kernel_launch(void* const* d_in, const int* in_sizes, int n_in,
                              void* d_out, int out_size, void* d_ws, size_t ws_size,
                              hipStream_t stream) {
    const float* x       = (const float*)d_in[0];
    const float* table   = (const float*)d_in[1];
    const int*   species = (const int*)d_in[2];
    const float* W0 = (const float*)d_in[3];
    const float* W1 = (const float*)d_in[4];
    const float* W2 = (const float*)d_in[5];
    const float* W3 = (const float*)d_in[6];
    const float* W4 = (const float*)d_in[7];

    float* out    = (float*)d_out;
    float* pc     = out;             /* [64,4,64,25] */
    float* logits = out + PC_SIZE;   /* [64,64,90,180] */
    float* ws     = (float*)d_ws;

    k_init<<<(WS_ZERO_N + 255) / 256, 256, 0, stream>>>(ws);
    k_nodes<<<1, 128, 0, stream>>>(ws);
    k_ybasis<<<(NT * 16 + 255) / 256, 256, 0, stream>>>(ws);

    dim3 gA(NG, LMAXV + 1);
    k_coeffs<<<gA, 256, 0, stream>>>(x, table, species, W0, W1, W2, W3, W4, pc, ws);

    dim3 gB(NT_CHUNKS, MT / 8);  /* 16 x 128 blocks, 8 waves/block */
    k_s2lse<<<gB, 256, 0, stream>>>(ws, logits);

    k_submax<<<2048, 256, 0, stream>>>(logits, ws);
}